// EdgeToNodeAggregation_188978561192
// MI455X (gfx1250) — compile-verified
//
#include <hip/hip_runtime.h>
#include <hip/hip_bf16.h>
#include <math.h>

#define N_NODES   100000
#define N_EDGES   500000
#define IN_DIM    16
#define HEADS     4
#define OUT_DIM   128
#define TOT_DIM   512           // HEADS * OUT_DIM
#define NEG_SLOPE 0.2f

typedef float v2f __attribute__((ext_vector_type(2)));
typedef float v8f __attribute__((ext_vector_type(8)));

// ---------------------------------------------------------------------------
// float atomic max via integer compare trick (works for all finite values,
// init value must be -inf = 0xFF800000)
// ---------------------------------------------------------------------------
__device__ __forceinline__ float atomicMaxF(float* addr, float val) {
    if (val >= 0.0f)
        return __int_as_float(atomicMax((int*)addr, __float_as_int(val)));
    else
        return __uint_as_float(atomicMin((unsigned int*)addr, __float_as_uint(val)));
}

__device__ __forceinline__ float leaky(float m) {
    return (m > 0.0f) ? m : m * NEG_SLOPE;
}

// ---------------------------------------------------------------------------
// Kernel 0: initialize out = bias (broadcast, float4), node_max = -inf,
// denom = 0.  N*TOT_DIM/4 = 12.8M threads covers everything.
// ---------------------------------------------------------------------------
__global__ __launch_bounds__(256)
void gat_init_kernel(float4* __restrict__ out4, const float4* __restrict__ bias4,
                     float* __restrict__ node_max, float* __restrict__ denom)
{
    int idx = blockIdx.x * blockDim.x + threadIdx.x;
    if (idx < N_NODES * (TOT_DIM / 4)) out4[idx] = bias4[idx & (TOT_DIM / 4 - 1)];
    if (idx < N_NODES * HEADS) {
        node_max[idx] = -INFINITY;
        denom[idx]    = 0.0f;
    }
}

// ---------------------------------------------------------------------------
// Kernel 1: fused dual GEMM via V_WMMA_F32_16X16X4_F32 (codegen-verified).
// One wave computes one 16x16 tile of XL (wave < TM*TN) or XR (otherwise).
//   XL = X @ W_l + b_l ; XR = X @ W_r + b_r       (K=16 -> 4 WMMA steps)
// A layout (16x4 f32): lanes 0-15 hold K={0,1}, lanes 16-31 hold K={2,3}.
// B layout (4x16 f32): VGPR v, lane-half g -> K = v + 2*g, N = lane%16.
// C/D layout: VGPR v, lane-half g -> M = v + 8*g, N = lane%16.
// ---------------------------------------------------------------------------
__global__ __launch_bounds__(256)
void gat_gemm_wmma_kernel(const float* __restrict__ X,
                          const float* __restrict__ Wl, const float* __restrict__ bl,
                          const float* __restrict__ Wr, const float* __restrict__ br,
                          float* __restrict__ XL, float* __restrict__ XR)
{
    const int TM = N_NODES / 16;   // 6250 (exact)
    const int TN = TOT_DIM / 16;   // 32   (exact)
    const int wave = (blockIdx.x * blockDim.x + threadIdx.x) >> 5;
    const int lane = threadIdx.x & 31;
    if (wave >= 2 * TM * TN) return;

    const int which = (wave >= TM * TN);
    const int t  = which ? wave - TM * TN : wave;
    const int tm = t / TN;
    const int tn = t % TN;
    const float* __restrict__ W = which ? Wr : Wl;
    const float* __restrict__ b = which ? br : bl;
    float* __restrict__ O       = which ? XR : XL;

    const int lmod  = lane & 15;
    const int lhalf = lane >> 4;          // 0 or 1
    const int row = tm * 16 + lmod;       // A: M = lane%16
    const int col = tn * 16 + lmod;       // B/D: N = lane%16
    const int khi = lhalf * 2;            // per-lane K sub-offset

    const float* __restrict__ xrow = X + row * IN_DIM;

    v8f c = {};
#pragma unroll
    for (int kc = 0; kc < 4; ++kc) {
        const int kbase = kc * 4 + khi;
        v2f a;
        a.x = xrow[kbase];
        a.y = xrow[kbase + 1];
        v2f bb;
        bb.x = W[(kbase)     * TOT_DIM + col];
        bb.y = W[(kbase + 1) * TOT_DIM + col];
        c = __builtin_amdgcn_wmma_f32_16x16x4_f32(
                /*neg_a=*/false, a, /*neg_b=*/false, bb,
                /*c_mod=*/(short)0, c, /*reuse_a=*/false, /*reuse_b=*/false);
    }

    const float bias = b[col];
    const int mbase = lhalf * 8;
#pragma unroll
    for (int v = 0; v < 8; ++v) {
        O[(size_t)(tm * 16 + mbase + v) * TOT_DIM + col] = c[v] + bias;
    }
}

// ---------------------------------------------------------------------------
// Kernel 2: per-(edge,head) GATv2 score + segment max.
// One wave32 per (edge, head); each lane loads exactly one float4 of xl, xr,
// att (global_load_b128, coalesced: 128 channels = 32 lanes * 4), then
// wave32 shuffle reduction; lane 0 writes e and atomic-maxes node_max[dst].
// ---------------------------------------------------------------------------
__global__ __launch_bounds__(256)
void gat_edge_score_kernel(const int* __restrict__ src, const int* __restrict__ dst,
                           const float* __restrict__ XL, const float* __restrict__ XR,
                           const float* __restrict__ att,
                           float* __restrict__ e, float* __restrict__ node_max)
{
    const int wid  = (blockIdx.x * blockDim.x + threadIdx.x) >> 5; // edge*HEADS + h
    if (wid >= N_EDGES * HEADS) return;
    const int lane = threadIdx.x & 31;
    const int edge = wid >> 2;
    const int h    = wid & 3;
    const int j = src[edge];
    const int i = dst[edge];

    const float4* __restrict__ xl4 = (const float4*)(XL + (size_t)j * TOT_DIM + h * OUT_DIM);
    const float4* __restrict__ xr4 = (const float4*)(XR + (size_t)i * TOT_DIM + h * OUT_DIM);
    const float4* __restrict__ a4  = (const float4*)(att + h * OUT_DIM);

    const float4 xl = xl4[lane];
    const float4 xr = xr4[lane];
    const float4 a  = a4[lane];

    float s = leaky(xl.x + xr.x) * a.x
            + leaky(xl.y + xr.y) * a.y
            + leaky(xl.z + xr.z) * a.z
            + leaky(xl.w + xr.w) * a.w;

#pragma unroll
    for (int off = 16; off > 0; off >>= 1)
        s += __shfl_down(s, off, 32);

    if (lane == 0) {
        e[wid] = s;
        atomicMaxF(&node_max[i * HEADS + h], s);
    }
}

// ---------------------------------------------------------------------------
// Kernel 3: ex = exp(e - max[dst]); denom[dst] += ex  (thread per edge-head)
// ---------------------------------------------------------------------------
__global__ __launch_bounds__(256)
void gat_edge_exp_kernel(const int* __restrict__ dst,
                         const float* __restrict__ node_max,
                         float* __restrict__ e, float* __restrict__ denom)
{
    int idx = blockIdx.x * blockDim.x + threadIdx.x;
    if (idx >= N_EDGES * HEADS) return;
    int edge = idx >> 2;
    int h    = idx & 3;
    int i    = dst[edge];
    float ex = expf(e[idx] - node_max[i * HEADS + h]);
    e[idx] = ex;                       // store numerator in-place
    atomicAdd(&denom[i * HEADS + h], ex);
}

// ---------------------------------------------------------------------------
// Kernel 4: out[dst] += alpha * xl[src].  One wave32 per edge.  Each of the
// 4 head-chunks (128 channels) is covered by one float4 per lane
// (global_load_b128); the 4 channel-atomics per chunk stay scalar f32
// (global_atomic_add_f32 without return).
// ---------------------------------------------------------------------------
__global__ __launch_bounds__(256)
void gat_aggregate_kernel(const int* __restrict__ src, const int* __restrict__ dst,
                          const float* __restrict__ XL,
                          const float* __restrict__ e, const float* __restrict__ denom,
                          float* __restrict__ out)
{
    const int edge = (blockIdx.x * blockDim.x + threadIdx.x) >> 5;
    if (edge >= N_EDGES) return;
    const int lane = threadIdx.x & 31;
    const int j = src[edge];
    const int i = dst[edge];

    const float4* __restrict__ xl4 = (const float4*)(XL + (size_t)j * TOT_DIM);
    float* __restrict__ op         = out + (size_t)i * TOT_DIM;

#pragma unroll
    for (int h = 0; h < HEADS; ++h) {
        const float al = e[edge * HEADS + h] / denom[i * HEADS + h];
        const float4 v = xl4[h * 32 + lane];
        const int k = h * OUT_DIM + lane * 4;
        atomicAdd(&op[k + 0], al * v.x);
        atomicAdd(&op[k + 1], al * v.y);
        atomicAdd(&op[k + 2], al * v.z);
        atomicAdd(&op[k + 3], al * v.w);
    }
}

// ---------------------------------------------------------------------------
// Host-side launch
// ---------------------------------------------------------------------------
extern "C" void kernel_launch(void* const* d_in, const int* in_sizes, int n_in,
                              void* d_out, int out_size, void* d_ws, size_t ws_size,
                              hipStream_t stream)
{
    const float* X    = (const float*)d_in[0];                 // [N, 16]
    const int*   eidx = (const int*)  d_in[1];                 // [2, E]
    const float* Wl   = (const float*)d_in[2];                 // [16, 512]
    const float* bl   = (const float*)d_in[3];                 // [512]
    const float* Wr   = (const float*)d_in[4];                 // [16, 512]
    const float* br   = (const float*)d_in[5];                 // [512]
    const float* att  = (const float*)d_in[6];                 // [4, 128]
    const float* bias = (const float*)d_in[7];                 // [512]
    float*       out  = (float*)d_out;                         // [N, 512]

    const int* src = eidx;              // row 0: message source j
    const int* dst = eidx + N_EDGES;    // row 1: aggregation target i

    // Workspace layout
    char* ws = (char*)d_ws;
    const size_t XL_bytes = (size_t)N_NODES * TOT_DIM * sizeof(float); // 204.8 MB
    float* XL       = (float*)(ws);
    float* XR       = (float*)(ws + XL_bytes);
    float* e        = (float*)(ws + 2 * XL_bytes);                               // E*H
    float* node_max = (float*)(ws + 2 * XL_bytes + (size_t)N_EDGES * HEADS * 4); // N*H
    float* denom    = (float*)(ws + 2 * XL_bytes + (size_t)N_EDGES * HEADS * 4
                                                 + (size_t)N_NODES * HEADS * 4); // N*H

    // 0) init (float4 stores for out)
    {
        int threads = N_NODES * (TOT_DIM / 4);                 // 12.8M
        gat_init_kernel<<<(threads + 255) / 256, 256, 0, stream>>>(
            (float4*)out, (const float4*)bias, node_max, denom);
    }
    // 1) dual WMMA GEMM (one wave per 16x16 tile, XL tiles then XR tiles)
    {
        int waves = 2 * (N_NODES / 16) * (TOT_DIM / 16);       // 400,000
        gat_gemm_wmma_kernel<<<(waves * 32 + 255) / 256, 256, 0, stream>>>(
            X, Wl, bl, Wr, br, XL, XR);
    }
    // 2) edge scores + segment max (one wave per (edge, head))
    {
        long long waves = (long long)N_EDGES * HEADS;          // 2,000,000
        gat_edge_score_kernel<<<(int)((waves * 32 + 255) / 256), 256, 0, stream>>>(
            src, dst, XL, XR, att, e, node_max);
    }
    // 3) exp + segment sum (one thread per (edge, head))
    {
        int threads = N_EDGES * HEADS;
        gat_edge_exp_kernel<<<(threads + 255) / 256, 256, 0, stream>>>(
            dst, node_max, e, denom);
    }
    // 4) alpha-weighted scatter aggregation (one wave per edge)
    {
        long long waves = (long long)N_EDGES;                  // 500,000
        gat_aggregate_kernel<<<(int)((waves * 32 + 255) / 256), 256, 0, stream>>>(
            src, dst, XL, e, denom, out);
    }
}